// MDETM_46789373722788
// MI455X (gfx1250) — compile-verified
//
#include <hip/hip_runtime.h>
#include <math.h>

// ---- problem constants (match reference) ----
#define S_    4
#define K_    50
#define T_    30
#define R_    300
#define V_    10000
#define HT_   800
#define EH_   200
#define BSZ_  256
#define SK_   200          // S_*K_
#define DELTA_ 0.005f
#define MAXLS  10.0f
#define MINLS  (-10.0f)

#define CDIV(a,b) (((a)+(b)-1)/(b))

typedef float v2f __attribute__((ext_vector_type(2)));
typedef float v8f __attribute__((ext_vector_type(8)));

__device__ __forceinline__ int clampi(int v, int hi) { return v < hi ? v : hi; }

#define WMMA_F32(a, b, c) \
    __builtin_amdgcn_wmma_f32_16x16x4_f32(false, (a), false, (b), (short)0, (c), false, false)

__device__ __forceinline__ void store_tile(float* __restrict__ C, int M, int N,
                                           int r0, int c0, const v8f acc,
                                           const float* __restrict__ bias, int relu,
                                           int hi, int l15)
{
    int c = c0 + l15;
#pragma unroll
    for (int i = 0; i < 8; i++) {
        int r = r0 + i + hi * 8;
        if (r < M && c < N) {
            float v = acc[i];
            if (bias) v += bias[c];
            if (relu) v = fmaxf(v, 0.f);
            C[(size_t)r * N + c] = v;
        }
    }
}

// =====================================================================
// WMMA f32 GEMM, "NT": C[M,N] = A(MxK, K contiguous) * B(NxK)^T
// WG = 8 waves arranged 4(M) x 2(N); each wave owns 32(M) x 64(N):
// 2 A-fragments x 4 B-fragments -> 8 accumulators, 8 WMMA per 6 loads.
// Fragments loaded and consumed in the same iteration (no cross-iter
// rotation -> no spills); uniform scalar loop control keeps EXEC all-1.
// =====================================================================
__global__ void gemm_f32_wmma_nt(const float* __restrict__ A,
                                 const float* __restrict__ B,
                                 float* __restrict__ C,
                                 int M, int N, int Kd,
                                 const float* __restrict__ bias, int relu)
{
    const int lane = threadIdx.x & 31;
    const int wave = threadIdx.x >> 5;
    const int l15  = lane & 15;
    const int hi   = lane >> 4;                  // K-pair half: {0,1} vs {2,3}
    const int m0 = blockIdx.x * 128 + (wave >> 1) * 32;
    const int n0 = blockIdx.y * 128 + (wave & 1) * 64;

    const float* Ap0 = A + (size_t)clampi(m0      + l15, M - 1) * Kd;
    const float* Ap1 = A + (size_t)clampi(m0 + 16 + l15, M - 1) * Kd;
    const float* Bp0 = B + (size_t)clampi(n0      + l15, N - 1) * Kd;
    const float* Bp1 = B + (size_t)clampi(n0 + 16 + l15, N - 1) * Kd;
    const float* Bp2 = B + (size_t)clampi(n0 + 32 + l15, N - 1) * Kd;
    const float* Bp3 = B + (size_t)clampi(n0 + 48 + l15, N - 1) * Kd;

    __builtin_prefetch(Ap0, 0, 1);               // global_prefetch_b8
    __builtin_prefetch(Bp0, 0, 1);

    const int ko = 2 * hi;
    v8f c00 = {0.f,0.f,0.f,0.f,0.f,0.f,0.f,0.f};
    v8f c01 = c00, c02 = c00, c03 = c00;
    v8f c10 = c00, c11 = c00, c12 = c00, c13 = c00;

    for (int kk = 0; kk < Kd; kk += 4) {         // uniform loop control
        const int k = kk + ko;
        v2f a0 = *(const v2f*)(Ap0 + k);         // 8B aligned: Kd even, k even
        v2f a1 = *(const v2f*)(Ap1 + k);
        v2f b0 = *(const v2f*)(Bp0 + k);
        v2f b1 = *(const v2f*)(Bp1 + k);
        v2f b2 = *(const v2f*)(Bp2 + k);
        v2f b3 = *(const v2f*)(Bp3 + k);
        c00 = WMMA_F32(a0, b0, c00);
        c01 = WMMA_F32(a0, b1, c01);
        c02 = WMMA_F32(a0, b2, c02);
        c03 = WMMA_F32(a0, b3, c03);
        c10 = WMMA_F32(a1, b0, c10);
        c11 = WMMA_F32(a1, b1, c11);
        c12 = WMMA_F32(a1, b2, c12);
        c13 = WMMA_F32(a1, b3, c13);
    }

    store_tile(C, M, N, m0,      n0,      c00, bias, relu, hi, l15);
    store_tile(C, M, N, m0,      n0 + 16, c01, bias, relu, hi, l15);
    store_tile(C, M, N, m0,      n0 + 32, c02, bias, relu, hi, l15);
    store_tile(C, M, N, m0,      n0 + 48, c03, bias, relu, hi, l15);
    store_tile(C, M, N, m0 + 16, n0,      c10, bias, relu, hi, l15);
    store_tile(C, M, N, m0 + 16, n0 + 16, c11, bias, relu, hi, l15);
    store_tile(C, M, N, m0 + 16, n0 + 32, c12, bias, relu, hi, l15);
    store_tile(C, M, N, m0 + 16, n0 + 48, c13, bias, relu, hi, l15);
}

// =====================================================================
// WMMA f32 GEMM, "NN": C[M,N] = A(MxK, K contiguous) * B(KxN, row-major)
// Same 128x128 WG tiling; B fragments via strided scalar loads.
// =====================================================================
__device__ __forceinline__ v2f load_b_nn(const float* __restrict__ B, int N,
                                         int k, int bc)
{
    v2f b;
    b.x = B[(size_t)k * N + bc];
    b.y = B[(size_t)(k + 1) * N + bc];
    return b;
}

__global__ void gemm_f32_wmma_nn(const float* __restrict__ A,
                                 const float* __restrict__ B,
                                 float* __restrict__ C,
                                 int M, int N, int Kd,
                                 const float* __restrict__ bias, int relu)
{
    const int lane = threadIdx.x & 31;
    const int wave = threadIdx.x >> 5;
    const int l15  = lane & 15;
    const int hi   = lane >> 4;
    const int m0 = blockIdx.x * 128 + (wave >> 1) * 32;
    const int n0 = blockIdx.y * 128 + (wave & 1) * 64;

    const float* Ap0 = A + (size_t)clampi(m0      + l15, M - 1) * Kd;
    const float* Ap1 = A + (size_t)clampi(m0 + 16 + l15, M - 1) * Kd;
    const int bc0 = clampi(n0      + l15, N - 1);
    const int bc1 = clampi(n0 + 16 + l15, N - 1);
    const int bc2 = clampi(n0 + 32 + l15, N - 1);
    const int bc3 = clampi(n0 + 48 + l15, N - 1);

    __builtin_prefetch(Ap0, 0, 1);

    const int ko = 2 * hi;
    v8f c00 = {0.f,0.f,0.f,0.f,0.f,0.f,0.f,0.f};
    v8f c01 = c00, c02 = c00, c03 = c00;
    v8f c10 = c00, c11 = c00, c12 = c00, c13 = c00;

    for (int kk = 0; kk < Kd; kk += 4) {
        const int k = kk + ko;
        v2f a0 = *(const v2f*)(Ap0 + k);
        v2f a1 = *(const v2f*)(Ap1 + k);
        v2f b0 = load_b_nn(B, N, k, bc0);
        v2f b1 = load_b_nn(B, N, k, bc1);
        v2f b2 = load_b_nn(B, N, k, bc2);
        v2f b3 = load_b_nn(B, N, k, bc3);
        c00 = WMMA_F32(a0, b0, c00);
        c01 = WMMA_F32(a0, b1, c01);
        c02 = WMMA_F32(a0, b2, c02);
        c03 = WMMA_F32(a0, b3, c03);
        c10 = WMMA_F32(a1, b0, c10);
        c11 = WMMA_F32(a1, b1, c11);
        c12 = WMMA_F32(a1, b2, c12);
        c13 = WMMA_F32(a1, b3, c13);
    }

    store_tile(C, M, N, m0,      n0,      c00, bias, relu, hi, l15);
    store_tile(C, M, N, m0,      n0 + 16, c01, bias, relu, hi, l15);
    store_tile(C, M, N, m0,      n0 + 32, c02, bias, relu, hi, l15);
    store_tile(C, M, N, m0,      n0 + 48, c03, bias, relu, hi, l15);
    store_tile(C, M, N, m0 + 16, n0,      c10, bias, relu, hi, l15);
    store_tile(C, M, N, m0 + 16, n0 + 16, c11, bias, relu, hi, l15);
    store_tile(C, M, N, m0 + 16, n0 + 32, c12, bias, relu, hi, l15);
    store_tile(C, M, N, m0 + 16, n0 + 48, c13, bias, relu, hi, l15);
}

// =====================================================================
// init: zero accumulators/flag, build src_mask from sources
// =====================================================================
__global__ void init_kernel(const int* __restrict__ sources,
                            float* __restrict__ acc, int* __restrict__ flag,
                            float* __restrict__ src_mask)
{
    int tid = threadIdx.x;
    if (tid < 4) acc[tid] = 0.f;
    if (tid == 4) *flag = 0;
    if (tid >= 8 && tid < 8 + S_) {
        int s = tid - 8;
        float m = 0.f;
        for (int b = 0; b < BSZ_; b++)
            if (sources[b] == s) m = 1.f;
        src_mask[s] = m;
    }
}

// =====================================================================
// kl_alpha: reduce over (S,K,T,R)
// =====================================================================
__global__ void kl_alpha_kernel(const float* __restrict__ mu,
                                const float* __restrict__ ls,
                                float* __restrict__ acc)
{
    __shared__ float red[256];
    const float log_delta = logf(DELTA_);
    const size_t total = (size_t)S_ * K_ * T_ * R_;
    float local = 0.f;
    for (size_t i = (size_t)blockIdx.x * blockDim.x + threadIdx.x; i < total;
         i += (size_t)gridDim.x * blockDim.x) {
        int t = (int)((i / R_) % T_);
        float m = mu[i], l = ls[i];
        if (t == 0) {
            local += 0.5f * ((expf(l) + m * m) / (1.0f + 1e-6f) - 1.0f - l);
        } else {
            float d = m - mu[i - R_];
            local += 0.5f * ((expf(l) + d * d) / (DELTA_ + 1e-6f) - 1.0f + log_delta - l);
        }
    }
    red[threadIdx.x] = local; __syncthreads();
    for (int s = 128; s > 0; s >>= 1) {
        if (threadIdx.x < s) red[threadIdx.x] += red[threadIdx.x + s];
        __syncthreads();
    }
    if (threadIdx.x == 0) atomicAdd(&acc[1], red[0]);
}

// =====================================================================
// Sequential LSTM (T steps) + eta chain in one workgroup.
// =====================================================================
__global__ void lstm_eta_kernel(const float* __restrict__ xWih,
                                const float* __restrict__ W_hh,
                                const float* __restrict__ b_hh,
                                const float* __restrict__ W_mu,
                                const float* __restrict__ b_mu,
                                const float* __restrict__ W_ls,
                                const float* __restrict__ b_ls,
                                float* __restrict__ etas,
                                float* __restrict__ acc)
{
    __shared__ float out_s[T_ * EH_];
    __shared__ float h[EH_], c[EH_], g[4 * EH_];
    __shared__ float eta_prev[SK_], mu_s[SK_], ls_s[SK_];
    __shared__ float red[256];
    __shared__ int   flag_s;
    __shared__ float kl_sh;
    const int tid = threadIdx.x;
    if (tid < EH_) { h[tid] = 0.f; c[tid] = 0.f; }
    if (tid == 0) kl_sh = 0.f;
    __syncthreads();

    // ---- LSTM ----
    for (int t = 0; t < T_; t++) {
        for (int j = tid; j < 4 * EH_; j += blockDim.x) {
            float s = xWih[t * 4 * EH_ + j] + b_hh[j];
            for (int e = 0; e < EH_; e++) s += h[e] * W_hh[e * 4 * EH_ + j];
            g[j] = s;
        }
        __syncthreads();
        for (int j = tid; j < EH_; j += blockDim.x) {
            float ig = 1.f / (1.f + expf(-g[j]));
            float fg = 1.f / (1.f + expf(-g[EH_ + j]));
            float gg = tanhf(g[2 * EH_ + j]);
            float og = 1.f / (1.f + expf(-g[3 * EH_ + j]));
            float cn = fg * c[j] + ig * gg;
            c[j] = cn;
            float hn = og * tanhf(cn);
            h[j] = hn;
            out_s[t * EH_ + j] = hn;
        }
        __syncthreads();
    }

    // ---- eta chain ----
    const float log_delta = logf(DELTA_);
    float local = 0.f;
    for (int j = tid; j < SK_; j += blockDim.x) {      // t = 0 (eta part is zero)
        float mu = b_mu[j], ls = b_ls[j];
        for (int e = 0; e < EH_; e++) {
            float oe = out_s[e];
            mu += oe * W_mu[e * SK_ + j];
            ls += oe * W_ls[e * SK_ + j];
        }
        eta_prev[j] = mu; etas[j] = mu;
        local += 0.5f * ((expf(ls) + mu * mu) / (1.f + 1e-6f) - 1.f - ls);
    }
    red[tid] = local; __syncthreads();
    for (int s2 = 128; s2 > 0; s2 >>= 1) { if (tid < s2) red[tid] += red[tid + s2]; __syncthreads(); }
    if (tid == 0) kl_sh += red[0];
    __syncthreads();

    for (int t = 1; t < T_; t++) {
        if (tid == 0) flag_s = 0;
        __syncthreads();
        for (int j = tid; j < SK_; j += blockDim.x) {
            float mu = b_mu[j], ls = b_ls[j];
            for (int e = 0; e < EH_; e++) {
                float oe = out_s[t * EH_ + e];
                float ep = eta_prev[e];                // SK_ == EH_
                mu += oe * W_mu[e * SK_ + j] + ep * W_mu[(EH_ + e) * SK_ + j];
                ls += oe * W_ls[e * SK_ + j] + ep * W_ls[(EH_ + e) * SK_ + j];
            }
            mu_s[j] = mu; ls_s[j] = ls;
            if (ls > MAXLS) atomicOr(&flag_s, 1);
        }
        __syncthreads();
        const int f = flag_s;
        local = 0.f;
        for (int j = tid; j < SK_; j += blockDim.x) {
            float ls = ls_s[j];
            ls = f ? fminf(ls, MAXLS) : fmaxf(ls, MINLS);   // _clamp_elif
            float mu = mu_s[j];
            float d = mu - eta_prev[j];
            local += 0.5f * ((expf(ls) + d * d) / (DELTA_ + 1e-6f) - 1.f + log_delta - ls);
            etas[t * SK_ + j] = mu;
        }
        red[tid] = local; __syncthreads();
        for (int s2 = 128; s2 > 0; s2 >>= 1) { if (tid < s2) red[tid] += red[tid + s2]; __syncthreads(); }
        if (tid == 0) kl_sh += red[0];
        __syncthreads();
        for (int j = tid; j < SK_; j += blockDim.x) eta_prev[j] = mu_s[j];
        __syncthreads();
    }
    if (tid == 0) atomicAdd(&acc[2], kl_sh);
}

// =====================================================================
// inp = concat(normalized_bows, etas[times]*mask_sk)  -> (BSZ, V+SK)
// =====================================================================
__global__ void build_inp_kernel(const float* __restrict__ nb,
                                 const int* __restrict__ times,
                                 const float* __restrict__ etas,
                                 const float* __restrict__ src_mask,
                                 float* __restrict__ inp)
{
    const int W = V_ + SK_;
    const size_t total = (size_t)BSZ_ * W;
    for (size_t i = (size_t)blockIdx.x * blockDim.x + threadIdx.x; i < total;
         i += (size_t)gridDim.x * blockDim.x) {
        int b = (int)(i / W);
        int j = (int)(i % W);
        float v;
        if (j < V_) v = nb[(size_t)b * V_ + j];
        else {
            int sk = j - V_;
            v = etas[times[b] * SK_ + sk] * src_mask[sk / K_];
        }
        inp[i] = v;
    }
}

// =====================================================================
// select mu/ls at (b, source_b) and compute global clamp flag
// =====================================================================
__global__ void select_flag_kernel(const float* __restrict__ mu_full,
                                   const float* __restrict__ ls_full,
                                   const int* __restrict__ sources,
                                   float* __restrict__ mu_sel,
                                   float* __restrict__ ls_sel,
                                   int* __restrict__ flag)
{
    int i = blockIdx.x * blockDim.x + threadIdx.x;
    if (i >= BSZ_ * K_) return;
    int b = i / K_, k = i % K_;
    int s = sources[b];
    float mu = mu_full[(size_t)b * SK_ + s * K_ + k];
    float ls = ls_full[(size_t)b * SK_ + s * K_ + k];
    mu_sel[i] = mu; ls_sel[i] = ls;
    if (ls > MAXLS) atomicOr(flag, 1);
}

// =====================================================================
// theta = softmax_K(mu_sel); kl_theta -> acc[3] (coeff applied at end)
// =====================================================================
__global__ void theta_kl_kernel(const float* __restrict__ mu_sel,
                                const float* __restrict__ ls_sel,
                                const int* __restrict__ flag,
                                const float* __restrict__ inp,
                                const int* __restrict__ sources,
                                float* __restrict__ theta,
                                float* __restrict__ acc)
{
    __shared__ float red[64];
    const int b = blockIdx.x, tid = threadIdx.x;
    const int s = sources[b];
    const int f = *flag;
    float mu = 0.f, ls = 0.f, eta = 0.f;
    if (tid < K_) {
        mu = mu_sel[b * K_ + tid];
        ls = ls_sel[b * K_ + tid];
        ls = f ? fminf(ls, MAXLS) : fmaxf(ls, MINLS);
        eta = inp[(size_t)b * (V_ + SK_) + V_ + s * K_ + tid];
    }
    red[tid] = (tid < K_) ? mu : -1e30f; __syncthreads();
    for (int s2 = 32; s2 > 0; s2 >>= 1) { if (tid < s2) red[tid] = fmaxf(red[tid], red[tid + s2]); __syncthreads(); }
    float mx = red[0]; __syncthreads();
    float e = (tid < K_) ? expf(mu - mx) : 0.f;
    red[tid] = e; __syncthreads();
    for (int s2 = 32; s2 > 0; s2 >>= 1) { if (tid < s2) red[tid] += red[tid + s2]; __syncthreads(); }
    float Z = red[0]; __syncthreads();
    if (tid < K_) theta[b * K_ + tid] = e / Z;
    float kl = 0.f;
    if (tid < K_) {
        float d = mu - eta;
        kl = 0.5f * ((expf(ls) + d * d) / (1.f + 1e-6f) - 1.f - ls);
    }
    red[tid] = kl; __syncthreads();
    for (int s2 = 32; s2 > 0; s2 >>= 1) { if (tid < s2) red[tid] += red[tid + s2]; __syncthreads(); }
    if (tid == 0) atomicAdd(&acc[3], red[0]);
}

// =====================================================================
// in-place softmax over each row of length V_  (logit -> beta)
// =====================================================================
__global__ void softmax_rows_kernel(float* __restrict__ logit)
{
    __shared__ float red[256];
    const int m = blockIdx.x, tid = threadIdx.x;
    float* row = logit + (size_t)m * V_;
    float mx = -1e30f;
    for (int v = tid; v < V_; v += 256) mx = fmaxf(mx, row[v]);
    red[tid] = mx; __syncthreads();
    for (int s = 128; s > 0; s >>= 1) { if (tid < s) red[tid] = fmaxf(red[tid], red[tid + s]); __syncthreads(); }
    mx = red[0]; __syncthreads();
    float sum = 0.f;
    for (int v = tid; v < V_; v += 256) { float e = expf(row[v] - mx); row[v] = e; sum += e; }
    red[tid] = sum; __syncthreads();
    for (int s = 128; s > 0; s >>= 1) { if (tid < s) red[tid] += red[tid + s]; __syncthreads(); }
    float inv = 1.f / red[0]; __syncthreads();
    for (int v = tid; v < V_; v += 256) row[v] *= inv;
}

// =====================================================================
// nll: per doc, mix_k theta * beta[s,k,t,:], -log * bows -> acc[0]
// =====================================================================
__global__ void nll_kernel(const float* __restrict__ beta,
                           const float* __restrict__ theta,
                           const float* __restrict__ bows,
                           const int* __restrict__ sources,
                           const int* __restrict__ times,
                           float* __restrict__ acc)
{
    __shared__ float th[K_];
    __shared__ float red[256];
    const int b = blockIdx.x, tid = threadIdx.x;
    const int s = sources[b], t = times[b];
    if (tid < K_) th[tid] = theta[b * K_ + tid];
    __syncthreads();
    const size_t base = ((size_t)s * K_ * T_ + t) * V_;   // k=0 row
    float local = 0.f;
    for (int v = tid; v < V_; v += 256) {
        float mix = 0.f;
        for (int k = 0; k < K_; k++)
            mix += th[k] * beta[base + (size_t)k * T_ * V_ + v];
        local += -logf(mix) * bows[(size_t)b * V_ + v];
    }
    red[tid] = local; __syncthreads();
    for (int s2 = 128; s2 > 0; s2 >>= 1) { if (tid < s2) red[tid] += red[tid + s2]; __syncthreads(); }
    if (tid == 0) atomicAdd(&acc[0], red[0]);
}

__global__ void finalize_kernel(const float* __restrict__ acc,
                                const int* __restrict__ num_docs,
                                float* __restrict__ out)
{
    if (threadIdx.x == 0 && blockIdx.x == 0) {
        float coeff = (float)num_docs[0] / (float)BSZ_;
        out[0] = acc[0] * coeff;   // nll
        out[1] = acc[1];           // kl_alpha
        out[2] = acc[2];           // kl_eta
        out[3] = acc[3] * coeff;   // kl_theta
    }
}

// =====================================================================
extern "C" void kernel_launch(void* const* d_in, const int* in_sizes, int n_in,
                              void* d_out, int out_size, void* d_ws, size_t ws_size,
                              hipStream_t stream)
{
    (void)in_sizes; (void)n_in; (void)out_size; (void)ws_size;
    // inputs (setup_inputs order)
    const float* bows      = (const float*)d_in[1];
    const float* nbows     = (const float*)d_in[2];
    const int*   times     = (const int*)d_in[3];
    const int*   sources   = (const int*)d_in[4];
    const float* rnn_inp   = (const float*)d_in[5];
    const int*   num_docs  = (const int*)d_in[6];
    const float* rho       = (const float*)d_in[7];
    const float* mu_qa     = (const float*)d_in[8];
    const float* ls_qa     = (const float*)d_in[9];
    const float* W_qt1     = (const float*)d_in[10];
    const float* b_qt1     = (const float*)d_in[11];
    const float* W_qt2     = (const float*)d_in[12];
    const float* b_qt2     = (const float*)d_in[13];
    const float* W_mu_th   = (const float*)d_in[14];
    const float* b_mu_th   = (const float*)d_in[15];
    const float* W_ls_th   = (const float*)d_in[16];
    const float* b_ls_th   = (const float*)d_in[17];
    const float* W_map     = (const float*)d_in[18];
    const float* b_map     = (const float*)d_in[19];
    const float* W_ih      = (const float*)d_in[20];
    const float* W_hh      = (const float*)d_in[21];
    const float* b_ih      = (const float*)d_in[22];
    const float* b_hh      = (const float*)d_in[23];
    const float* W_mu_eta  = (const float*)d_in[24];
    const float* b_mu_eta  = (const float*)d_in[25];
    const float* W_ls_eta  = (const float*)d_in[26];
    const float* b_ls_eta  = (const float*)d_in[27];

    // workspace carve-up (logit/beta buffer last: 240MB)
    char* w = (char*)d_ws;
    size_t off = 0;
    auto take = [&](size_t nbytes) -> void* {
        void* p = w + off;
        off = (off + nbytes + 255) & ~(size_t)255;
        return p;
    };
    float* acc      = (float*)take(4 * sizeof(float));
    int*   flag     = (int*)take(sizeof(int));
    float* src_mask = (float*)take(S_ * sizeof(float));
    float* x        = (float*)take((size_t)T_ * EH_ * 4);
    float* xWih     = (float*)take((size_t)T_ * 4 * EH_ * 4);
    float* etas     = (float*)take((size_t)T_ * SK_ * 4);
    float* inp      = (float*)take((size_t)BSZ_ * (V_ + SK_) * 4);
    float* h1       = (float*)take((size_t)BSZ_ * HT_ * 4);
    float* h2       = (float*)take((size_t)BSZ_ * HT_ * 4);
    float* mu_full  = (float*)take((size_t)BSZ_ * SK_ * 4);
    float* ls_full  = (float*)take((size_t)BSZ_ * SK_ * 4);
    float* mu_sel   = (float*)take((size_t)BSZ_ * K_ * 4);
    float* ls_sel   = (float*)take((size_t)BSZ_ * K_ * 4);
    float* theta    = (float*)take((size_t)BSZ_ * K_ * 4);
    float* logit    = (float*)take((size_t)S_ * K_ * T_ * V_ * 4);   // becomes beta

    init_kernel<<<1, 64, 0, stream>>>(sources, acc, flag, src_mask);
    kl_alpha_kernel<<<1024, 256, 0, stream>>>(mu_qa, ls_qa, acc);

    // x = rnn_inp @ W_map + b_map          (30 x 200, K=10000)
    dim3 gx(CDIV(T_, 128), CDIV(EH_, 128));
    gemm_f32_wmma_nn<<<gx, 256, 0, stream>>>(rnn_inp, W_map, x, T_, EH_, V_, b_map, 0);
    // xWih = x @ W_ih + b_ih               (30 x 800, K=200)
    dim3 gi(CDIV(T_, 128), CDIV(4 * EH_, 128));
    gemm_f32_wmma_nn<<<gi, 256, 0, stream>>>(x, W_ih, xWih, T_, 4 * EH_, EH_, b_ih, 0);

    lstm_eta_kernel<<<1, 256, 0, stream>>>(xWih, W_hh, b_hh,
                                           W_mu_eta, b_mu_eta, W_ls_eta, b_ls_eta,
                                           etas, acc);

    build_inp_kernel<<<2048, 256, 0, stream>>>(nbows, times, etas, src_mask, inp);

    // h1 = relu(inp @ W_qt1 + b)           (256 x 800, K=10200)
    dim3 gq(CDIV(BSZ_, 128), CDIV(HT_, 128));
    gemm_f32_wmma_nn<<<gq, 256, 0, stream>>>(inp, W_qt1, h1, BSZ_, HT_, V_ + SK_, b_qt1, 1);
    // h2 = relu(h1 @ W_qt2 + b)            (256 x 800, K=800)
    gemm_f32_wmma_nn<<<gq, 256, 0, stream>>>(h1, W_qt2, h2, BSZ_, HT_, HT_, b_qt2, 1);
    // mu_th / ls_th                        (256 x 200, K=800)
    dim3 gt(CDIV(BSZ_, 128), CDIV(SK_, 128));
    gemm_f32_wmma_nn<<<gt, 256, 0, stream>>>(h2, W_mu_th, mu_full, BSZ_, SK_, HT_, b_mu_th, 0);
    gemm_f32_wmma_nn<<<gt, 256, 0, stream>>>(h2, W_ls_th, ls_full, BSZ_, SK_, HT_, b_ls_th, 0);

    select_flag_kernel<<<CDIV(BSZ_ * K_, 256), 256, 0, stream>>>(
        mu_full, ls_full, sources, mu_sel, ls_sel, flag);
    theta_kl_kernel<<<BSZ_, 64, 0, stream>>>(mu_sel, ls_sel, flag, inp, sources, theta, acc);

    // logit = alphas(6000x300) @ rho(10000x300)^T  — dominant GEMM, NT
    dim3 gl(CDIV(S_ * K_ * T_, 128), CDIV(V_, 128));
    gemm_f32_wmma_nt<<<gl, 256, 0, stream>>>(mu_qa, rho, logit,
                                             S_ * K_ * T_, V_, R_, nullptr, 0);
    softmax_rows_kernel<<<S_ * K_ * T_, 256, 0, stream>>>(logit);   // -> beta in place
    nll_kernel<<<BSZ_, 256, 0, stream>>>(logit, theta, bows, sources, times, acc);

    finalize_kernel<<<1, 32, 0, stream>>>(acc, num_docs, (float*)d_out);
}